// MultiBoxLoss_combined_67697274519899
// MI455X (gfx1250) — compile-verified
//
#include <hip/hip_runtime.h>
#include <math.h>

#define NWAY   5
#define NSHOT  5
#define NQUERY 8
#define NP     8732
#define NC     32
#define NO     4
#define NSIMG  (NWAY*NSHOT)     /* 25 */
#define NQIMG  (NWAY*NQUERY)    /* 40 */
#define NIMG   (NSIMG+NQIMG)    /* 65 */
#define NROWS  (NQIMG*NP)       /* 349280 */
#define NTILES (NROWS/16)       /* 21830, exact */
#define MK     9                /* ceil(NP/1024) */

typedef float v2f __attribute__((ext_vector_type(2)));
typedef float v8f __attribute__((ext_vector_type(8)));

#define SEL4(a,o) ((o)==0?(a)[0]:((o)==1?(a)[1]:((o)==2?(a)[2]:(a)[3])))

__device__ __forceinline__ float smoothl1(float d) {
  d = fabsf(d);
  return d < 1.f ? 0.5f * d * d : d - 0.5f;
}

// ---------------------------------------------------------------------------
// Kernel 1: SSD prior matching per image + inline smooth-L1 loc loss.
// One block per image (65). Exact argmax tie-breaking (first index) both ways.
// ---------------------------------------------------------------------------
__global__ __launch_bounds__(256) void match_kernel(
    const float* __restrict__ s_loc, const float* __restrict__ q_loc,
    const float* __restrict__ priors, const float* __restrict__ s_t,
    const float* __restrict__ q_t,
    float* __restrict__ conf_t, int* __restrict__ cnt_nz,
    int* __restrict__ cnt_pos, float* __restrict__ partial_l)
{
  __shared__ float bov[NP];
  __shared__ unsigned char bidx[NP];
  __shared__ float tb[NO][256];
  __shared__ int   tbi[NO][256];
  __shared__ float redf[256];
  __shared__ int   redi[256];

  const int img = blockIdx.x;
  const int tid = threadIdx.x;

  float tx0[NO], ty0[NO], tx1[NO], ty1[NO], lab[NO], tarea[NO];
  const float* locp;
  if (img < NSIMG) {
    const float* base = s_t + (size_t)img * NO * 5;
#pragma unroll
    for (int o = 0; o < NO; ++o) {
      tx0[o] = base[o*5+0]; ty0[o] = base[o*5+1];
      tx1[o] = base[o*5+2]; ty1[o] = base[o*5+3];
      lab[o] = base[o*5+4];
    }
    locp = s_loc + (size_t)img * NP * 4;
  } else {
    const int qi = img - NSIMG;
    const int w  = qi / NQUERY;
    const float* base = q_t + (size_t)qi * NO * 5;
#pragma unroll
    for (int o = 0; o < NO; ++o) {
      tx0[o] = base[o*5+0]; ty0[o] = base[o*5+1];
      tx1[o] = base[o*5+2]; ty1[o] = base[o*5+3];
      lab[o] = (base[o*5+4] == -1.0f) ? -1.0f : (float)(w + 1);
    }
    locp = q_loc + (size_t)qi * NP * 4;
  }
#pragma unroll
  for (int o = 0; o < NO; ++o)
    tarea[o] = (tx1[o]-tx0[o]) * (ty1[o]-ty0[o]);

  float mto[NO]; int mtp[NO];
#pragma unroll
  for (int o = 0; o < NO; ++o) { mto[o] = -1.f; mtp[o] = 0; }

  const float4* pr4 = (const float4*)priors;
  for (int p = tid; p < NP; p += 256) {
    float4 pr = pr4[p];
    float px0 = pr.x - pr.z*0.5f, py0 = pr.y - pr.w*0.5f;
    float px1 = pr.x + pr.z*0.5f, py1 = pr.y + pr.w*0.5f;
    float parea = (px1-px0)*(py1-py0);
    float bo = -1.f; int bi = 0;
#pragma unroll
    for (int o = 0; o < NO; ++o) {
      float ix0 = fmaxf(tx0[o], px0), iy0 = fmaxf(ty0[o], py0);
      float ix1 = fminf(tx1[o], px1), iy1 = fminf(ty1[o], py1);
      float iw = fmaxf(ix1-ix0, 0.f), ih = fmaxf(iy1-iy0, 0.f);
      float inter = iw*ih;
      float ov = inter / (tarea[o] + parea - inter);
      if (ov > bo) { bo = ov; bi = o; }          // first-max over truths
      if (ov > mto[o]) { mto[o] = ov; mtp[o] = p; } // first-max within thread
    }
    bov[p] = bo; bidx[p] = (unsigned char)bi;
  }
#pragma unroll
  for (int o = 0; o < NO; ++o) { tb[o][tid] = mto[o]; tbi[o][tid] = mtp[o]; }
  __syncthreads();
  if (tid == 0) {
    for (int o = 0; o < NO; ++o) {
      float best = -1.f; int bp = 0;
      for (int t = 0; t < 256; ++t) {
        float v = tb[o][t]; int i = tbi[o][t];
        if (v > best || (v == best && i < bp)) { best = v; bp = i; }
      }
      bov[bp] = 2.0f; bidx[bp] = (unsigned char)o;  // sequential: last truth wins
    }
  }
  __syncthreads();

  float locl = 0.f; int cnz = 0, cpos = 0;
  const float4* ld4 = (const float4*)locp;
  for (int p = tid; p < NP; p += 256) {
    float ovp = bov[p]; int o = (int)bidx[p];
    float cf = (ovp < 0.5f) ? 0.f : SEL4(lab, o);
    conf_t[(size_t)img * NP + p] = cf;
    if (cf != 0.f) {
      cnz++;
      if (cf > 0.f) cpos++;
      float mx0 = SEL4(tx0,o), my0 = SEL4(ty0,o);
      float mx1 = SEL4(tx1,o), my1 = SEL4(ty1,o);
      float4 pr = pr4[p];
      float gcx = ((mx0+mx1)*0.5f - pr.x) / (0.1f * pr.z);
      float gcy = ((my0+my1)*0.5f - pr.y) / (0.1f * pr.w);
      float gw  = logf((mx1-mx0) / pr.z) / 0.2f;
      float gh  = logf((my1-my0) / pr.w) / 0.2f;
      float4 ld = ld4[p];
      locl += smoothl1(ld.x - gcx) + smoothl1(ld.y - gcy)
            + smoothl1(ld.z - gw)  + smoothl1(ld.w - gh);
    }
  }
  redf[tid] = locl; __syncthreads();
  for (int s = 128; s > 0; s >>= 1) { if (tid < s) redf[tid] += redf[tid+s]; __syncthreads(); }
  if (tid == 0) partial_l[img] = redf[0];
  redi[tid] = cnz; __syncthreads();
  for (int s = 128; s > 0; s >>= 1) { if (tid < s) redi[tid] += redi[tid+s]; __syncthreads(); }
  if (tid == 0) cnt_nz[img] = redi[0];
  __syncthreads();
  redi[tid] = cpos; __syncthreads();
  for (int s = 128; s > 0; s >>= 1) { if (tid < s) redi[tid] += redi[tid+s]; __syncthreads(); }
  if (tid == 0) cnt_pos[img] = redi[0];
}

// ---------------------------------------------------------------------------
// Kernel 2: class prototypes (masked mean of support conf) + ||proto||^2.
// One block per way.
// ---------------------------------------------------------------------------
__global__ __launch_bounds__(256) void proto_kernel(
    const float* __restrict__ s_conf, const float* __restrict__ conf_t,
    const int* __restrict__ cnt_pos, float* __restrict__ proto,
    float* __restrict__ pp)
{
  const int w = blockIdx.x;
  const int c = threadIdx.x & 31;
  const int g = threadIdx.x >> 5;
  float acc = 0.f;
  for (int s = 0; s < NSHOT; ++s) {
    const int img = w * NSHOT + s;
    const float* cp = s_conf + (size_t)img * NP * NC;
    const float* ct = conf_t + (size_t)img * NP;
    for (int p = g; p < NP; p += 8) {
      float m = (ct[p] > 0.f) ? 1.f : 0.f;
      acc += cp[(size_t)p * NC + c] * m;
    }
  }
  __shared__ float buf[8][32];
  buf[g][c] = acc;
  __syncthreads();
  if (g == 0) {
    float t = buf[0][c];
    for (int k = 1; k < 8; ++k) t += buf[k][c];
    int cnt = 0;
    for (int s = 0; s < NSHOT; ++s) cnt += cnt_pos[w * NSHOT + s];
    float pv = t / (float)cnt;
    proto[w * NC + c] = pv;
    buf[0][c] = pv * pv;
  }
  __syncthreads();
  if (threadIdx.x == 0) {
    float s2 = 0.f;
    for (int k = 0; k < 32; ++k) s2 += buf[0][k];
    pp[w] = s2;
  }
}

// ---------------------------------------------------------------------------
// Kernel 3: query distances via V_WMMA_F32_16X16X4_F32 + fused obj/conf CE.
// ||q-p||^2 = ||q||^2 - 2 q.p + ||p||^2 ; cross term is a [16x32]x[32x16] GEMM
// per tile, done as 8 chained K=4 f32 WMMAs (exact f32 math).
// ---------------------------------------------------------------------------
__global__ __launch_bounds__(256) void qce_kernel(
    const float* __restrict__ q_conf, const float* __restrict__ q_obj,
    const float* __restrict__ conf_t, const float* __restrict__ proto,
    const float* __restrict__ pp, float* __restrict__ ce_c)
{
  __shared__ float protoPad[16 * NC];
  __shared__ float qcbuf[8][16][8];
  const int tid = threadIdx.x;
  for (int i = tid; i < 16 * NC; i += 256) {
    int n = i >> 5;
    protoPad[i] = (n < NWAY) ? proto[i] : 0.f;
  }
  __syncthreads();
  const int lane = tid & 31;
  const int wv   = tid >> 5;
  const int half = lane >> 4;
  const int n    = lane & 15;

  // B fragment (loop invariant): B[k][n] = proto[n][k], zero-padded n>=5.
  v2f b[8];
#pragma unroll
  for (int j = 0; j < 8; ++j) {
    int k = 4*j + 2*half;
    b[j][0] = protoPad[n * NC + k];
    b[j][1] = protoPad[n * NC + k + 1];
  }
  const float ppn = (n < NWAY) ? pp[n] : 0.f;

  const int gwv = blockIdx.x * 8 + wv;
  const int nwv = gridDim.x * 8;
  for (int tile = gwv; tile < NTILES; tile += nwv) {
    const int rowBase = tile * 16;
    const float* ap = q_conf + (size_t)(rowBase + n) * NC;
    if (tile + nwv < NTILES)
      __builtin_prefetch(ap + (size_t)nwv * 16 * NC, 0, 0);

    v2f a[8];
    float qpart = 0.f;
#pragma unroll
    for (int j = 0; j < 8; ++j) {
      int k = 4*j + 2*half;
      a[j] = *(const v2f*)(ap + k);
      qpart += a[j][0]*a[j][0] + a[j][1]*a[j][1];
    }
    // full row norm: lane L and L^16 hold complementary halves of row (L&15)
    float qq = qpart + __shfl_xor(qpart, 16, 32);
    float qrow[8];
#pragma unroll
    for (int r = 0; r < 8; ++r) qrow[r] = __shfl(qq, 8*half + r, 32);

    v8f cacc = {};
#pragma unroll
    for (int j = 0; j < 8; ++j)
      cacc = __builtin_amdgcn_wmma_f32_16x16x4_f32(
          false, a[j], false, b[j], (short)0, cacc, false, false);

    if (n < NWAY) {
#pragma unroll
      for (int r = 0; r < 8; ++r) {
        float d2 = qrow[r] + ppn - 2.f * cacc[r];
        qcbuf[wv][8*half + r][n] = -sqrtf(fmaxf(d2, 0.f));
      }
    }
    __asm__ volatile("" ::: "memory");   // keep LDS write->read order (in-order per wave)
    if (lane < 16) {
      const int row = rowBase + lane;
      float q0 = qcbuf[wv][lane][0], q1 = qcbuf[wv][lane][1],
            q2 = qcbuf[wv][lane][2], q3 = qcbuf[wv][lane][3],
            q4 = qcbuf[wv][lane][4];
      float m5 = fmaxf(fmaxf(fmaxf(q0,q1), fmaxf(q2,q3)), q4);
      float lse5 = logf(expf(q0-m5)+expf(q1-m5)+expf(q2-m5)+expf(q3-m5)+expf(q4-m5)) + m5;
      float o0 = q_obj[(size_t)row*2], o1 = q_obj[(size_t)row*2+1];
      float l0 = o0 + lse5, l1 = o1 + q0, l2 = o1 + q1,
            l3 = o1 + q2,   l4 = o1 + q3, l5 = o1 + q4;
      float tc = conf_t[(size_t)NSIMG * NP + row];
      int t = (tc == -1.f) ? 0 : (int)tc;
      float m6 = fmaxf(fmaxf(fmaxf(l0,l1), fmaxf(l2,l3)), fmaxf(l4,l5));
      float s6 = expf(l0-m6)+expf(l1-m6)+expf(l2-m6)+expf(l3-m6)+expf(l4-m6)+expf(l5-m6);
      float lt = l0;
      lt = (t==1)?l1:lt; lt = (t==2)?l2:lt; lt = (t==3)?l3:lt;
      lt = (t==4)?l4:lt; lt = (t==5)?l5:lt;
      ce_c[row] = logf(s6) + m6 - lt;
    }
    __asm__ volatile("" ::: "memory");
  }
}

// ---------------------------------------------------------------------------
// Kernel 4: conf hard-negative mining (exact stable rank via LDS scan) + sum.
// One block per query image.
// ---------------------------------------------------------------------------
__global__ __launch_bounds__(1024) void mine_conf_kernel(
    const float* __restrict__ ce_c, const float* __restrict__ conf_t,
    const int* __restrict__ cnt_pos, float* __restrict__ partial_c)
{
  __shared__ float mine[NP];
  __shared__ float red[1024];
  const int img = blockIdx.x;
  const int tid = threadIdx.x;
  const float* ct = conf_t + (size_t)(NSIMG + img) * NP;
  const float* ce = ce_c + (size_t)img * NP;
  for (int p = tid; p < NP; p += 1024) {
    float c = ct[p];
    mine[p] = (c != 0.f) ? 0.f : ce[p];
  }
  __syncthreads();
  int nn = 3 * cnt_pos[NSIMG + img];
  if (nn > NP - 1) nn = NP - 1;
  int pk[MK]; float vk[MK]; int gk[MK];
#pragma unroll
  for (int k = 0; k < MK; ++k) {
    pk[k] = tid + k * 1024;
    vk[k] = (pk[k] < NP) ? mine[pk[k]] : -1.f;
    gk[k] = 0;
  }
  for (int j = 0; j < NP; ++j) {
    float m = mine[j];
#pragma unroll
    for (int k = 0; k < MK; ++k)
      gk[k] += (int)((m > vk[k]) || (m == vk[k] && j < pk[k]));
  }
  float acc = 0.f;
#pragma unroll
  for (int k = 0; k < MK; ++k) {
    if (pk[k] < NP) {
      float c = ct[pk[k]];
      bool sel = ((c > 0.f) || (gk[k] < nn)) && (c >= 0.f);
      if (sel) acc += ce[pk[k]];
    }
  }
  red[tid] = acc; __syncthreads();
  for (int s = 512; s > 0; s >>= 1) { if (tid < s) red[tid] += red[tid+s]; __syncthreads(); }
  if (tid == 0) partial_c[img] = red[0];
}

// ---------------------------------------------------------------------------
// Kernel 5: objectness CE + hard-negative mining + sum. One block per image.
// ---------------------------------------------------------------------------
__global__ __launch_bounds__(1024) void mine_obj_kernel(
    const float* __restrict__ s_obj, const float* __restrict__ q_obj,
    const float* __restrict__ conf_t, const int* __restrict__ cnt_nz,
    float* __restrict__ partial_o)
{
  __shared__ float mine[NP];
  __shared__ float red[1024];
  const int img = blockIdx.x;
  const int tid = threadIdx.x;
  const float* op = (img < NSIMG) ? (s_obj + (size_t)img * NP * 2)
                                  : (q_obj + (size_t)(img - NSIMG) * NP * 2);
  const float* ct = conf_t + (size_t)img * NP;
  for (int p = tid; p < NP; p += 1024) {
    float o0 = op[(size_t)p*2], o1 = op[(size_t)p*2+1];
    bool t = (ct[p] != 0.f);
    float m = fmaxf(o0, o1);
    float cev = logf(expf(o0-m) + expf(o1-m)) + m - (t ? o1 : o0);
    mine[p] = t ? 0.f : cev;
  }
  __syncthreads();
  int nn = 3 * cnt_nz[img];
  if (nn > NP - 1) nn = NP - 1;
  int pk[MK]; float vk[MK]; int gk[MK];
#pragma unroll
  for (int k = 0; k < MK; ++k) {
    pk[k] = tid + k * 1024;
    vk[k] = (pk[k] < NP) ? mine[pk[k]] : -1.f;
    gk[k] = 0;
  }
  for (int j = 0; j < NP; ++j) {
    float m = mine[j];
#pragma unroll
    for (int k = 0; k < MK; ++k)
      gk[k] += (int)((m > vk[k]) || (m == vk[k] && j < pk[k]));
  }
  float acc = 0.f;
#pragma unroll
  for (int k = 0; k < MK; ++k) {
    if (pk[k] < NP) {
      bool t = (ct[pk[k]] != 0.f);
      if (t || (gk[k] < nn)) {
        float o0 = op[(size_t)pk[k]*2], o1 = op[(size_t)pk[k]*2+1];
        float m = fmaxf(o0, o1);
        acc += logf(expf(o0-m) + expf(o1-m)) + m - (t ? o1 : o0);
      }
    }
  }
  red[tid] = acc; __syncthreads();
  for (int s = 512; s > 0; s >>= 1) { if (tid < s) red[tid] += red[tid+s]; __syncthreads(); }
  if (tid == 0) partial_o[img] = red[0];
}

// ---------------------------------------------------------------------------
// Kernel 6: deterministic fixed-order final sums + normalization.
// ---------------------------------------------------------------------------
__global__ void finalize_kernel(
    const float* __restrict__ pl, const float* __restrict__ pc,
    const float* __restrict__ po, const int* __restrict__ cnz,
    const int* __restrict__ cpos, float* __restrict__ out)
{
  if (threadIdx.x == 0 && blockIdx.x == 0) {
    float ll = 0.f, lc = 0.f, lo = 0.f; int sq = 0, qn = 0;
    for (int i = 0; i < NIMG; ++i) { ll += pl[i]; lo += po[i]; sq += cnz[i]; }
    for (int i = 0; i < NQIMG; ++i) { lc += pc[i]; qn += cpos[NSIMG + i]; }
    out[0] = ll / (float)sq;
    out[1] = lc / (float)qn;
    out[2] = lo / (float)sq;
  }
}

extern "C" void kernel_launch(void* const* d_in, const int* in_sizes, int n_in,
                              void* d_out, int out_size, void* d_ws, size_t ws_size,
                              hipStream_t stream) {
  (void)in_sizes; (void)n_in; (void)out_size; (void)ws_size;
  const float* s_loc  = (const float*)d_in[0];
  const float* s_conf = (const float*)d_in[1];
  const float* s_obj  = (const float*)d_in[2];
  const float* q_loc  = (const float*)d_in[3];
  const float* q_conf = (const float*)d_in[4];
  const float* q_obj  = (const float*)d_in[5];
  const float* priors = (const float*)d_in[6];
  const float* s_t    = (const float*)d_in[7];
  const float* q_t    = (const float*)d_in[8];
  float* out = (float*)d_out;

  char* ws = (char*)d_ws;
  size_t off = 0;
  auto take = [&](size_t bytes) -> char* {
    char* p = ws + off;
    off = (off + bytes + 255) & ~(size_t)255;
    return p;
  };
  float* conf_t    = (float*)take((size_t)NIMG * NP * sizeof(float));
  float* ce_c      = (float*)take((size_t)NROWS * sizeof(float));
  float* proto     = (float*)take((size_t)NWAY * NC * sizeof(float));
  float* pp        = (float*)take((size_t)NWAY * sizeof(float));
  float* partial_l = (float*)take((size_t)NIMG * sizeof(float));
  float* partial_c = (float*)take((size_t)NQIMG * sizeof(float));
  float* partial_o = (float*)take((size_t)NIMG * sizeof(float));
  int*   cnt_nz    = (int*)take((size_t)NIMG * sizeof(int));
  int*   cnt_pos   = (int*)take((size_t)NIMG * sizeof(int));

  match_kernel<<<NIMG, 256, 0, stream>>>(s_loc, q_loc, priors, s_t, q_t,
                                         conf_t, cnt_nz, cnt_pos, partial_l);
  proto_kernel<<<NWAY, 256, 0, stream>>>(s_conf, conf_t, cnt_pos, proto, pp);
  qce_kernel<<<512, 256, 0, stream>>>(q_conf, q_obj, conf_t, proto, pp, ce_c);
  mine_conf_kernel<<<NQIMG, 1024, 0, stream>>>(ce_c, conf_t, cnt_pos, partial_c);
  mine_obj_kernel<<<NIMG, 1024, 0, stream>>>(s_obj, q_obj, conf_t, cnt_nz, partial_o);
  finalize_kernel<<<1, 32, 0, stream>>>(partial_l, partial_c, partial_o,
                                        cnt_nz, cnt_pos, out);
}